// DeformConv_89094801588658
// MI455X (gfx1250) — compile-verified
//
#include <hip/hip_runtime.h>

typedef __attribute__((ext_vector_type(2))) float v2f;
typedef __attribute__((ext_vector_type(8))) float v8f;

#define BATCH 4
#define CIN   64
#define COUT  64
#define HH    128
#define WW    128
#define HWSZ  (HH * WW)
#define KK2   9
#define OFFC  18            // 2 * K2 offset channels
#define SSTR  65            // padded LDS row stride (bank-conflict free)

// ---------------------------------------------------------------------------
// Kernel 1: 3x3 conv, 64 -> 18 channels (offset prediction), pad = 1.
// One thread per (b, oc, pixel). Lanes run along pixels -> coalesced loads.
// ---------------------------------------------------------------------------
__global__ __launch_bounds__(256) void offset_conv_kernel(
    const float* __restrict__ x, const float* __restrict__ w,
    const float* __restrict__ bias, float* __restrict__ offset)
{
    const int boc = blockIdx.y;            // b*18 + oc
    const int oc  = boc % OFFC;
    const int b   = boc / OFFC;
    const int p   = blockIdx.x * 256 + threadIdx.x;
    const int h   = p >> 7;
    const int wd  = p & (WW - 1);

    const float* xb = x + (size_t)b * CIN * HWSZ;
    const float* wc = w + (size_t)oc * CIN * 9;

    float acc = bias[oc];
    for (int c = 0; c < CIN; ++c) {
        const float* xc = xb + (size_t)c * HWSZ;
        const float* wk = wc + c * 9;
        #pragma unroll
        for (int ky = 0; ky < 3; ++ky) {
            const int yy = h + ky - 1;
            if (yy < 0 || yy >= HH) continue;
            #pragma unroll
            for (int kx = 0; kx < 3; ++kx) {
                const int xx = wd + kx - 1;
                if (xx < 0 || xx >= WW) continue;
                acc = fmaf(xc[yy * WW + xx], wk[ky * 3 + kx], acc);
            }
        }
    }
    offset[(size_t)boc * HWSZ + p] = acc;
}

// ---------------------------------------------------------------------------
// Kernel 2: deformable conv via fp32 WMMA (V_WMMA_F32_16X16X4_F32).
// Block = 128 threads (4 waves), owns 64 consecutive pixels of one batch.
// Per tap k: stage w[k][c][o] (64x64) in LDS, bilinear-sample 64px x 64c tile
// into LDS, then each wave runs a 16px x 64out GEMM with K=64 in chunks of 4.
// Accumulators (4 x v8f per lane) persist across all 9 taps.
// ---------------------------------------------------------------------------
__global__ __launch_bounds__(128) void deform_wmma_kernel(
    const float* __restrict__ x, const float* __restrict__ offset,
    const float* __restrict__ dw, float* __restrict__ out)
{
    __shared__ float s_lds[CIN * SSTR];     // sampled [c][pl], reused as [o][pl]
    __shared__ float w_lds[CIN * COUT];     // weights [c][o] for current tap
    __shared__ float o_lds[OFFC * 64];      // offsets [ch][pl]

    const int t    = threadIdx.x;
    const int lane = t & 31;
    const int wave = t >> 5;                // 0..3
    const int tile = blockIdx.x;            // 1024 tiles
    const int b    = tile >> 8;             // 256 pixel-tiles per batch
    const int p0   = (tile & 255) * 64;     // base linear pixel

    // ---- stage offsets for this pixel tile: 18 ch x 64 px ----
    for (int i = t; i < OFFC * 64; i += 128) {
        const int ch = i >> 6;
        const int pl = i & 63;
        o_lds[i] = offset[((size_t)(b * OFFC + ch)) * HWSZ + p0 + pl];
    }

    // sampling-phase mapping: thread -> (pixel, c-half)
    const int pl_me = t & 63;
    const int chalf = t >> 6;               // 0/1 -> c in [0,32) or [32,64)
    const int p  = p0 + pl_me;
    const int h  = p >> 7;
    const int wd = p & (WW - 1);
    const float* xb = x + (size_t)b * CIN * HWSZ;

    // GEMM-phase fragment mapping (ISA 7.12.2 layouts)
    const int m_a   = lane & 15;            // M (A) / N (B) index
    const int pl_a  = wave * 16 + m_a;      // this wave's pixel for row M
    const int khalf = (lane >> 4) << 1;     // K base offset: 0 or 2

    v8f acc[4];
    const v8f vzero = {0.f, 0.f, 0.f, 0.f, 0.f, 0.f, 0.f, 0.f};
    #pragma unroll
    for (int i = 0; i < 4; ++i) acc[i] = vzero;

    for (int k = 0; k < KK2; ++k) {
        __syncthreads();   // previous GEMM done reading s_lds / w_lds

        // ---- stage weight slice: w_lds[c][o] = dw[o][c][k] ----
        for (int i = t; i < CIN * COUT; i += 128) {
            const int c = i >> 6;
            const int o = i & 63;
            w_lds[i] = dw[(size_t)o * (CIN * KK2) + c * KK2 + k];
        }

        // ---- bilinear sampling for this tap ----
        const int   ky  = k / 3;
        const int   kx  = k % 3;
        const float dy  = o_lds[(2 * k    ) * 64 + pl_me];
        const float dx  = o_lds[(2 * k + 1) * 64 + pl_me];
        const float py  = (float)h + (float)(ky - 1) + dy;
        const float px  = (float)wd + (float)(kx - 1) + dx;
        const float y0f = floorf(py);
        const float x0f = floorf(px);
        const int   y0  = (int)y0f;
        const int   x0  = (int)x0f;
        const float wy1 = py - y0f, wx1 = px - x0f;
        const float wy0 = 1.f - wy1, wx0 = 1.f - wx1;
        const bool  vy0 = (y0     >= 0) && (y0     < HH);
        const bool  vy1 = (y0 + 1 >= 0) && (y0 + 1 < HH);
        const bool  vx0 = (x0     >= 0) && (x0     < WW);
        const bool  vx1 = (x0 + 1 >= 0) && (x0 + 1 < WW);
        const float f00 = (vy0 && vx0) ? wy0 * wx0 : 0.f;
        const float f01 = (vy0 && vx1) ? wy0 * wx1 : 0.f;
        const float f10 = (vy1 && vx0) ? wy1 * wx0 : 0.f;
        const float f11 = (vy1 && vx1) ? wy1 * wx1 : 0.f;
        const int yc0 = min(max(y0,     0), HH - 1);
        const int yc1 = min(max(y0 + 1, 0), HH - 1);
        const int xc0 = min(max(x0,     0), WW - 1);
        const int xc1 = min(max(x0 + 1, 0), WW - 1);
        const int i00 = yc0 * WW + xc0;
        const int i01 = yc0 * WW + xc1;
        const int i10 = yc1 * WW + xc0;
        const int i11 = yc1 * WW + xc1;

        #pragma unroll 8
        for (int ci = 0; ci < 32; ++ci) {
            const int c = chalf * 32 + ci;
            const float* xc = xb + (size_t)c * HWSZ;
            const float s = f00 * xc[i00] + f01 * xc[i01] +
                            f10 * xc[i10] + f11 * xc[i11];
            s_lds[c * SSTR + pl_me] = s;
        }

        __syncthreads();   // staging complete

        // ---- GEMM: 16px x 64out, K = 64 channels in chunks of 4 ----
        #pragma unroll 4
        for (int cc = 0; cc < CIN; cc += 4) {
            const int c0 = cc + khalf;
            v2f a;
            a.x = s_lds[(c0    ) * SSTR + pl_a];
            a.y = s_lds[(c0 + 1) * SSTR + pl_a];
            #pragma unroll
            for (int ot = 0; ot < 4; ++ot) {
                v2f bf;
                bf.x = w_lds[(c0    ) * 64 + ot * 16 + m_a];
                bf.y = w_lds[(c0 + 1) * 64 + ot * 16 + m_a];
                acc[ot] = __builtin_amdgcn_wmma_f32_16x16x4_f32(
                    false, a, false, bf, (short)0, acc[ot], false, false);
            }
        }
    }

    // ---- transpose through LDS so stores are coalesced ----
    __syncthreads();
    #pragma unroll
    for (int ot = 0; ot < 4; ++ot) {
        const int o = ot * 16 + (lane & 15);
        #pragma unroll
        for (int v = 0; v < 8; ++v) {
            const int m = v + ((lane >> 4) << 3);
            s_lds[o * SSTR + wave * 16 + m] = acc[ot][v];
        }
    }
    __syncthreads();
    for (int i = t; i < COUT * 64; i += 128) {
        const int o  = i >> 6;
        const int pl = i & 63;
        out[((size_t)(b * COUT + o)) * HWSZ + p0 + pl] = s_lds[o * SSTR + pl];
    }
}

// ---------------------------------------------------------------------------
extern "C" void kernel_launch(void* const* d_in, const int* in_sizes, int n_in,
                              void* d_out, int out_size, void* d_ws, size_t ws_size,
                              hipStream_t stream) {
    const float* x        = (const float*)d_in[0];  // (4,64,128,128)
    const float* offset_w = (const float*)d_in[1];  // (18,64,3,3)
    const float* offset_b = (const float*)d_in[2];  // (18,)
    const float* deform_w = (const float*)d_in[3];  // (64,64,3,3)
    float*       out      = (float*)d_out;          // (4,64,128,128)
    float*       offs     = (float*)d_ws;           // (4,18,128,128) scratch

    dim3 g1(HWSZ / 256, BATCH * OFFC);
    offset_conv_kernel<<<g1, 256, 0, stream>>>(x, offset_w, offset_b, offs);

    const int tiles = BATCH * (HWSZ / 64);          // 1024 blocks
    deform_wmma_kernel<<<tiles, 128, 0, stream>>>(x, offs, deform_w, out);
}